// MovementEmbeddingModule_55920474194510
// MI455X (gfx1250) — compile-verified
//
#include <hip/hip_runtime.h>
#include <cstdint>

namespace {
constexpr int BS = 4, D = 16, K = 10, C = 3, H = 128, W = 128;
constexpr int HW = H * W;               // 16384 pixels
constexpr int IMG_FLOATS = C * HW;      // 49152 floats = 192 KB in LDS
constexpr int PLANE = D * HW;           // channel stride in output (262144)
constexpr int NTHR = 512;               // 16 wave32 waves per block
}

typedef float v4f __attribute__((ext_vector_type(4)));

__launch_bounds__(NTHR, 1)
__global__ void movement_embedding_kernel(
    const float* __restrict__ kp_app,   // (4,1,10,2)
    const float* __restrict__ kp_vid,   // (4,16,10,2)
    const float* __restrict__ img,      // (4,3,1,128,128)
    float* __restrict__ out)            // (4,60,16,128,128)
{
  __shared__ float simg[IMG_FLOATS];    // planar 3x128x128 image for batch b
  __shared__ float stab[4 * 128];       // [ex1 | ex0 | ey1 | ey0] separable Gaussians

  const int blk = blockIdx.x;           // one block per (b, d, k)
  const int b  = blk / (D * K);
  const int r  = blk - b * (D * K);
  const int dd = r / K;
  const int kk = r - dd * K;
  const int tid = threadIdx.x;

  // ---- Stage the 3-channel image for batch b into LDS via the gfx1250
  //      async global->LDS DMA path (ASYNCcnt). 24 full-EXEC b128 rounds. ----
  {
    const float* gsrc = img + (size_t)b * IMG_FLOATS;   // contiguous 192 KB
    const uint32_t lds0 = (uint32_t)(uintptr_t)(void*)&simg[0];
    for (int i = tid; i < IMG_FLOATS / 4; i += NTHR) {
      uint32_t laddr = lds0 + (uint32_t)i * 16u;
      uint64_t gaddr = (uint64_t)(uintptr_t)(gsrc + (size_t)i * 4);
      asm volatile("global_load_async_to_lds_b128 %0, %1, off"
                   :: "v"(laddr), "v"(gaddr) : "memory");
    }
  }

  // ---- Per-(b,d,k) scalars (mirror reference math) ----
  const size_t vidBase  = ((size_t)b * D * K + (size_t)dd * K + kk) * 2;
  const size_t vid0Base = ((size_t)b * D * K + kk) * 2;
  const size_t appBase  = ((size_t)b * K + kk) * 2;
  const float vx  = kp_vid[vidBase + 0],  vy  = kp_vid[vidBase + 1];
  const float v0x = kp_vid[vid0Base + 0], v0y = kp_vid[vid0Base + 1];
  const float ax  = kp_app[appBase + 0],  ay  = kp_app[appBase + 1];
  const float dfx = vx - v0x, dfy = vy - v0y;   // diff
  const float mx  = dfx + ax, my  = dfy + ay;   // kp_mean
  const float vdx = -dfx,     vdy = -dfy;       // kp_video_diff

  // ---- Separable Gaussian tables: one __expf per thread ----
  {
    const int j = tid & 127;
    const int which = tid >> 7;                 // 0:ex1(mx) 1:ex0(ax) 2:ey1(my) 3:ey0(ay)
    const float gc = (float)j * (2.0f / 127.0f) - 1.0f;
    const float mm = (which == 0) ? mx : (which == 1) ? ax : (which == 2) ? my : ay;
    const float dv = gc - mm;
    stab[tid] = __expf(-50.0f * dv * dv);       // -0.5 / KP_VARIANCE
  }

  // ---- Block-uniform bilinear shift: integer offset + constant weights ----
  const float sx = vdx * 63.5f;                 // (W-1)/2
  const float sy = vdy * 63.5f;
  const float fsx = floorf(sx), fsy = floorf(sy);
  const int   sxi = (int)fsx,   syi = (int)fsy;
  const float wx1 = sx - fsx, wx0 = 1.0f - wx1;
  const float wy1 = sy - fsy, wy0 = 1.0f - wy1;
  const float w00 = wx0 * wy0, w10 = wx1 * wy0;
  const float w01 = wx0 * wy1, w11 = wx1 * wy1;

  asm volatile("s_wait_asynccnt 0" ::: "memory");
  __syncthreads();   // LDS image + tables visible to all waves

  float* outb = out + (size_t)b * 60 * PLANE + (size_t)(kk * 6) * PLANE
                    + (size_t)dd * HW;
  const v4f cvdx = {vdx, vdx, vdx, vdx};
  const v4f cvdy = {vdy, vdy, vdy, vdy};

  // 4-pixel groups: 4096 groups / 512 threads = 8 iterations
  for (int it = 0; it < HW / (4 * NTHR); ++it) {
    const int g  = tid + it * NTHR;             // group id
    const int y  = g >> 5;                      // 32 groups per row
    const int x0 = (g & 31) << 2;

    const float ey1 = stab[256 + y], ey0 = stab[384 + y];
    const int iy0 = y + syi, iy1 = iy0 + 1;
    const bool vy0f = (unsigned)iy0 < (unsigned)H;
    const bool vy1f = (unsigned)iy1 < (unsigned)H;
    const int ry0 = (iy0 & (H - 1)) << 7;
    const int ry1 = (iy1 & (H - 1)) << 7;

    v4f fh, f0, f1, f2;
#pragma unroll
    for (int j = 0; j < 4; ++j) {
      const int x = x0 + j;
      fh[j] = stab[x] * ey1 - stab[128 + x] * ey0;  // heat(d) - heat(0)

      const int ix0 = x + sxi, ix1 = ix0 + 1;
      const bool vx0f = (unsigned)ix0 < (unsigned)W;
      const bool vx1f = (unsigned)ix1 < (unsigned)W;
      const int cx0 = ix0 & (W - 1), cx1 = ix1 & (W - 1);
      const float m00 = (vx0f && vy0f) ? w00 : 0.0f;
      const float m10 = (vx1f && vy0f) ? w10 : 0.0f;
      const float m01 = (vx0f && vy1f) ? w01 : 0.0f;
      const float m11 = (vx1f && vy1f) ? w11 : 0.0f;
      const int o00 = ry0 + cx0, o10 = ry0 + cx1;
      const int o01 = ry1 + cx0, o11 = ry1 + cx1;

      f0[j] = simg[o00] * m00 + simg[o10] * m10
            + simg[o01] * m01 + simg[o11] * m11;
      f1[j] = simg[HW + o00] * m00 + simg[HW + o10] * m10
            + simg[HW + o01] * m01 + simg[HW + o11] * m11;
      f2[j] = simg[2 * HW + o00] * m00 + simg[2 * HW + o10] * m10
            + simg[2 * HW + o01] * m01 + simg[2 * HW + o11] * m11;
    }

    const size_t base = (size_t)y * W + x0;     // 16B-aligned
    __builtin_nontemporal_store(fh,   (v4f*)(outb + base));
    __builtin_nontemporal_store(cvdx, (v4f*)(outb + (size_t)PLANE + base));
    __builtin_nontemporal_store(cvdy, (v4f*)(outb + (size_t)2 * PLANE + base));
    __builtin_nontemporal_store(f0,   (v4f*)(outb + (size_t)3 * PLANE + base));
    __builtin_nontemporal_store(f1,   (v4f*)(outb + (size_t)4 * PLANE + base));
    __builtin_nontemporal_store(f2,   (v4f*)(outb + (size_t)5 * PLANE + base));
  }
}

extern "C" void kernel_launch(void* const* d_in, const int* in_sizes, int n_in,
                              void* d_out, int out_size, void* d_ws, size_t ws_size,
                              hipStream_t stream) {
  (void)in_sizes; (void)n_in; (void)out_size; (void)d_ws; (void)ws_size;
  const float* kp_app = (const float*)d_in[0];   // kp_appearance_mean
  const float* kp_vid = (const float*)d_in[1];   // kp_video_mean
  const float* img    = (const float*)d_in[2];   // appearance_frame
  float* out = (float*)d_out;
  movement_embedding_kernel<<<BS * D * K, NTHR, 0, stream>>>(kp_app, kp_vid, img, out);
}